// SAGENet_11338713661814
// MI455X (gfx1250) — compile-verified
//
#include <hip/hip_runtime.h>
#include <math.h>

#define N_NODES   1000000
#define C_IN      128
#define HID       32
#define OUT_F     16
#define NSRC1     600000
#define NDST1     80000
#define E1        1280000
#define NDST2     8000
#define E2        80000

typedef float v2f __attribute__((ext_vector_type(2)));
typedef float v8f __attribute__((ext_vector_type(8)));

#define LWT_STRIDE 132   // transposed W1 tile: lane stride 132 ≡ 4 banks, halves offset ≡ 2 -> conflict-free
#define W2T_STRIDE 34

// ---------------------------------------------------------------------------
// Kernel 1: z1[s,:] = x[n_id[s],:] @ W1   (f32 WMMA 16x16x4, M=16 / N=32 per wave)
// ---------------------------------------------------------------------------
__global__ __launch_bounds__(256)
void gemm1_gather_wmma(const float* __restrict__ x,
                       const int* __restrict__ n_id,
                       const float* __restrict__ W1,
                       float* __restrict__ z1) {
    __shared__ float lwT[HID * LWT_STRIDE];       // W1 transposed: lwT[n][k], ~16.9 KB
    for (int i = threadIdx.x; i < C_IN * HID; i += blockDim.x) {
        const int k = i >> 5, n = i & 31;
        lwT[n * LWT_STRIDE + k] = W1[i];
    }
    __syncthreads();

    const int wave = threadIdx.x >> 5;
    const int lane = threadIdx.x & 31;
    const int half = lane >> 4;        // 0: K=k..k+1,  1: K=k+2..k+3
    const int mrow = lane & 15;        // A: M index,  B/D: N index

    const int tile = blockIdx.x * 8 + wave;       // wave-uniform guard (EXEC stays all-1s)
    if (tile * 16 >= NSRC1) return;
    const int row0 = tile * 16;

    const size_t arow = (size_t)n_id[row0 + mrow] * C_IN;

    v8f c0 = {}; v8f c1 = {};
    for (int k = 0; k < C_IN; k += 4) {
        const int kk = k + 2 * half;
        const v2f a   = *(const v2f*)(x + arow + kk);                      // b64, 8B aligned
        const v2f b0  = *(const v2f*)&lwT[mrow * LWT_STRIDE + kk];         // ds_load_b64
        const v2f b1v = *(const v2f*)&lwT[(16 + mrow) * LWT_STRIDE + kk];  // ds_load_b64
        c0 = __builtin_amdgcn_wmma_f32_16x16x4_f32(false, a, false, b0,  (short)0, c0, false, false);
        c1 = __builtin_amdgcn_wmma_f32_16x16x4_f32(false, a, false, b1v, (short)0, c1, false, false);
    }
    #pragma unroll
    for (int i = 0; i < 8; ++i) {
        const int m = row0 + 8 * half + i;        // D layout: lane half selects M block
        z1[(size_t)m * HID + mrow]      = c0[i];
        z1[(size_t)m * HID + 16 + mrow] = c1[i];
    }
}

// ---------------------------------------------------------------------------
// Kernel 2: per-destination degree histogram (1 int atomic per edge)
// ---------------------------------------------------------------------------
__global__ __launch_bounds__(256)
void hist1(const int* __restrict__ dst1, int* __restrict__ cnt1) {
    const int e = blockIdx.x * blockDim.x + threadIdx.x;
    if (e >= E1) return;
    atomicAdd(&cnt1[dst1[e]], 1);
}

// ---------------------------------------------------------------------------
// Kernel 3: exclusive prefix scan over 80k degrees (single workgroup)
// ---------------------------------------------------------------------------
__global__ __launch_bounds__(1024)
void scan_rowstart(const int* __restrict__ cnt1, int* __restrict__ rowstart) {
    __shared__ int tmp[1024];
    __shared__ int sbase;
    if (threadIdx.x == 0) sbase = 0;
    __syncthreads();
    for (int chunk = 0; chunk < NDST1; chunk += 1024) {
        const int i = chunk + threadIdx.x;
        const int v = (i < NDST1) ? cnt1[i] : 0;
        tmp[threadIdx.x] = v;
        __syncthreads();
        for (int off = 1; off < 1024; off <<= 1) {      // Kogge-Stone inclusive
            int t = 0;
            if (threadIdx.x >= off) t = tmp[threadIdx.x - off];
            __syncthreads();
            tmp[threadIdx.x] += t;
            __syncthreads();
        }
        const int incl = tmp[threadIdx.x];
        const int base = sbase;
        if (i < NDST1) rowstart[i] = base + incl - v;   // exclusive
        __syncthreads();
        if (threadIdx.x == 1023) sbase = base + incl;
        __syncthreads();
    }
    if (threadIdx.x == 0) rowstart[NDST1] = sbase;      // == E1
}

// ---------------------------------------------------------------------------
// Kernel 4: bucket fill -> CSR edge list by destination (1 int atomic per edge)
// ---------------------------------------------------------------------------
__global__ __launch_bounds__(256)
void fill1(const int* __restrict__ src1, const int* __restrict__ dst1,
           const int* __restrict__ rowstart, int* __restrict__ cursor,
           int* __restrict__ ebuf) {
    const int e = blockIdx.x * blockDim.x + threadIdx.x;
    if (e >= E1) return;
    const int d = dst1[e];
    const int pos = rowstart[d] + atomicAdd(&cursor[d], 1);
    ebuf[pos] = src1[e];
}

// ---------------------------------------------------------------------------
// Kernel 5: gather-form mean + bias + ReLU (NO atomics). wave per dst, lane=feat
// ---------------------------------------------------------------------------
__global__ __launch_bounds__(256)
void gather_mean_relu(const int* __restrict__ rowstart, const int* __restrict__ ebuf,
                      const float* __restrict__ z1, const float* __restrict__ b1,
                      float* __restrict__ h1) {
    const int d    = (blockIdx.x * blockDim.x + threadIdx.x) >> 5;
    const int lane = threadIdx.x & 31;
    if (d >= NDST1) return;
    const int beg = rowstart[d];
    const int end = rowstart[d + 1];

    float acc = 0.0f;
    for (int base = beg; base < end; base += 32) {
        const int rem = end - base;
        const int n = rem < 32 ? rem : 32;
        int myidx = 0;
        if (lane < n) myidx = ebuf[base + lane];        // one coalesced index load / 32 edges
        for (int j = 0; j < n; ++j) {
            const int s = __shfl(myidx, j, 32);         // broadcast edge src
            acc += z1[(size_t)s * HID + lane];          // 128B coalesced, L2-resident
        }
    }
    const float inv = 1.0f / fmaxf((float)(end - beg), 1.0f);
    h1[(size_t)d * HID + lane] = fmaxf(fmaf(acc, inv, b1[lane]), 0.0f);
}

// ---------------------------------------------------------------------------
// Kernel 6: y2 = h1 @ W2   (f32 WMMA 16x16x4, K=32, N=16)
// ---------------------------------------------------------------------------
__global__ __launch_bounds__(256)
void gemm2_wmma(const float* __restrict__ h1, const float* __restrict__ W2,
                float* __restrict__ y2) {
    __shared__ float w2t[OUT_F * W2T_STRIDE];     // W2 transposed: w2t[n][k]
    for (int i = threadIdx.x; i < HID * OUT_F; i += blockDim.x) {
        const int k = i >> 4, n = i & 15;
        w2t[n * W2T_STRIDE + k] = W2[i];
    }
    __syncthreads();

    const int wave = threadIdx.x >> 5;
    const int lane = threadIdx.x & 31;
    const int half = lane >> 4;
    const int mrow = lane & 15;

    const int tile = blockIdx.x * 8 + wave;
    if (tile * 16 >= NDST1) return;
    const int row0 = tile * 16;
    const int r    = row0 + mrow;

    v8f c = {};
    #pragma unroll
    for (int k = 0; k < HID; k += 4) {
        const int kk = k + 2 * half;
        const v2f a = *(const v2f*)&h1[(size_t)r * HID + kk];       // contiguous b64
        const v2f b = *(const v2f*)&w2t[mrow * W2T_STRIDE + kk];    // ds_load_b64
        c = __builtin_amdgcn_wmma_f32_16x16x4_f32(false, a, false, b, (short)0, c, false, false);
    }
    #pragma unroll
    for (int i = 0; i < 8; ++i) {
        const int m = row0 + 8 * half + i;
        y2[(size_t)m * OUT_F + mrow] = c[i];
    }
}

// ---------------------------------------------------------------------------
// Kernel 7: hop-2 edge scatter (small: ~1.3M atomics total)
// ---------------------------------------------------------------------------
__global__ __launch_bounds__(256)
void scatter_add2(const int* __restrict__ src2, const int* __restrict__ dst2,
                  const float* __restrict__ y2,
                  float* __restrict__ acc2, float* __restrict__ cnt2) {
    const int idx = blockIdx.x * blockDim.x + threadIdx.x;
    const int e = idx >> 4;
    const int f = idx & 15;
    if (e >= E2) return;
    const int s = src2[e];
    const int d = dst2[e];
    atomicAdd(&acc2[(size_t)d * OUT_F + f], y2[(size_t)s * OUT_F + f]);
    if (f == 0) atomicAdd(&cnt2[d], 1.0f);
}

// ---------------------------------------------------------------------------
// Kernel 8: out = log_softmax(acc2/cnt + b2), one thread per destination row
// ---------------------------------------------------------------------------
__global__ __launch_bounds__(256)
void logsoftmax_out(const float* __restrict__ acc2, const float* __restrict__ cnt2,
                    const float* __restrict__ b2, float* __restrict__ out) {
    const int d = blockIdx.x * blockDim.x + threadIdx.x;
    if (d >= NDST2) return;
    const float inv = 1.0f / fmaxf(cnt2[d], 1.0f);
    float v[OUT_F];
    float m = -INFINITY;
    #pragma unroll
    for (int j = 0; j < OUT_F; ++j) {
        v[j] = fmaf(acc2[(size_t)d * OUT_F + j], inv, b2[j]);
        m = fmaxf(m, v[j]);
    }
    float s = 0.0f;
    #pragma unroll
    for (int j = 0; j < OUT_F; ++j) s += expf(v[j] - m);
    const float lse = m + logf(s);
    #pragma unroll
    for (int j = 0; j < OUT_F; ++j) out[(size_t)d * OUT_F + j] = v[j] - lse;
}

// ---------------------------------------------------------------------------
extern "C" void kernel_launch(void* const* d_in, const int* in_sizes, int n_in,
                              void* d_out, int out_size, void* d_ws, size_t ws_size,
                              hipStream_t stream) {
    const float* x    = (const float*)d_in[0];
    const int*   n_id = (const int*)  d_in[1];
    const int*   src1 = (const int*)  d_in[2];
    const int*   dst1 = (const int*)  d_in[3];
    const int*   src2 = (const int*)  d_in[4];
    const int*   dst2 = (const int*)  d_in[5];
    const float* W1   = (const float*)d_in[6];
    const float* b1   = (const float*)d_in[7];
    const float* W2   = (const float*)d_in[8];
    const float* b2   = (const float*)d_in[9];
    float* out = (float*)d_out;

    // Workspace layout (4-byte units); everything needing zeroing is contiguous.
    float* ws       = (float*)d_ws;
    float* z1       = ws;                                  // NSRC1*HID
    float* h1       = z1 + (size_t)NSRC1 * HID;            // NDST1*HID
    float* y2       = h1 + (size_t)NDST1 * HID;            // NDST1*OUT_F
    int*   ebuf     = (int*)(y2 + (size_t)NDST1 * OUT_F);  // E1
    int*   rowstart = ebuf + E1;                           // NDST1+1
    int*   cnt1     = rowstart + (NDST1 + 1);              // NDST1  <- zeroed region start
    int*   cursor   = cnt1 + NDST1;                        // NDST1
    float* acc2     = (float*)(cursor + NDST1);            // NDST2*OUT_F
    float* cnt2     = acc2 + (size_t)NDST2 * OUT_F;        // NDST2

    const size_t zero_bytes =
        ((size_t)NDST1 * 2 + (size_t)NDST2 * OUT_F + NDST2) * 4;
    hipMemsetAsync(cnt1, 0, zero_bytes, stream);

    gemm1_gather_wmma<<<(NSRC1 / 16 + 7) / 8, 256, 0, stream>>>(x, n_id, W1, z1);
    hist1            <<<(E1 + 255) / 256, 256, 0, stream>>>(dst1, cnt1);
    scan_rowstart    <<<1, 1024, 0, stream>>>(cnt1, rowstart);
    fill1            <<<(E1 + 255) / 256, 256, 0, stream>>>(src1, dst1, rowstart, cursor, ebuf);
    gather_mean_relu <<<(NDST1 * 32 + 255) / 256, 256, 0, stream>>>(rowstart, ebuf, z1, b1, h1);
    gemm2_wmma       <<<(NDST1 / 16 + 7) / 8, 256, 0, stream>>>(h1, W2, y2);
    scatter_add2     <<<((size_t)E2 * 16 + 255) / 256, 256, 0, stream>>>(src2, dst2, y2, acc2, cnt2);
    logsoftmax_out   <<<(NDST2 + 255) / 256, 256, 0, stream>>>(acc2, cnt2, b2, out);
}